// REM_12463995093375
// MI455X (gfx1250) — compile-verified
//
#include <hip/hip_runtime.h>
#include <stdint.h>

#define TRUNC_ 2048
#define TLEN_  2048
#define CAP_   200
#define NH_    8

// ---------------------------------------------------------------------------
// Workspace layout (16 KB):
//   ws[h*512 +       d] = TA[h][d]   d = 0..255  (only 0..200 ever read)
//   ws[h*512 + 256 + d] = TB[h][d]
//
// Per output head h, for j <= i:
//   d  = i - j           ; L  = (d  > 200) ? 0 : d
//   d2 = i/n - j/n       ; L2 = (d2 > 200) ? 0 : d2   (n = 1 except h2:4, h6:3, h7:2)
//   val = (i%n == j%n) ? TA[L2] * TB[L] : 0           (TB == 1 for h0..h5)
//   if (i == j) val -= 1
// ---------------------------------------------------------------------------

__global__ void build_tables_kernel(const float* __restrict__ eta,
                                    const float* __restrict__ nu,
                                    const float* __restrict__ theta,
                                    float* __restrict__ ws)
{
    int t = blockIdx.x * blockDim.x + threadIdx.x;   // 0 .. 4095
    if (t >= NH_ * 2 * 256) return;
    int h   = t >> 9;
    int sel = (t >> 8) & 1;
    int d   = t & 255;
    float fd = (float)d;
    float v;
    if (sel == 0) {
        // TA
        if (h < 3) {
            // ipow(tanh(eta_h), d)  -- sign-correct for negative base
            float lam = tanhf(eta[h]);
            float m = powf(fabsf(lam), fd);
            if ((d & 1) && lam < 0.f) m = -m;
            v = m;
        } else if (h < 6) {
            // gam^d * trig(theta*d), gam = sigmoid(nu) > 0
            int g = h - 3;
            float gam = 1.f / (1.f + expf(-nu[g]));
            float m = powf(gam, fd);
            float tr = (g < 2) ? cosf(theta[g] * fd) : sinf(theta[g] * fd);
            v = m * tr;
        } else if (h == 6) {
            v = cosf(theta[3] * fd);
        } else {
            v = sinf(theta[4] * fd);
        }
    } else {
        // TB
        if (h < 6)       v = 1.f;
        else if (h == 6) v = cosf(theta[3] * fd);
        else             v = sinf(theta[4] * fd);
    }
    ws[t] = v;
}

template<int N>
__device__ __forceinline__ void emit_row(const float* __restrict__ sTA,
                                         const float* __restrict__ sTB,
                                         float* __restrict__ outRow,
                                         int i)
{
    const int j0 = threadIdx.x * 8;
    float r[8];
#pragma unroll
    for (int k = 0; k < 8; ++k) {
        int j = j0 + k;
        float v = 0.f;
        if (j <= i) {
            int d = i - j;
            int L = (d > CAP_) ? 0 : d;
            if (N == 1) {
                // TB == 1 for all n==1 heads: single LDS read per element
                v = sTA[L];
            } else {
                if ((i % N) == (j % N)) {
                    int d2 = i / N - j / N;
                    int L2 = (d2 > CAP_) ? 0 : d2;
                    v = sTA[L2] * sTB[L];
                }
            }
            if (d == 0) v -= 1.f;
        }
        r[k] = v;
    }
    float4* o = (float4*)(outRow + j0);
    o[0] = make_float4(r[0], r[1], r[2], r[3]);
    o[1] = make_float4(r[4], r[5], r[6], r[7]);
}

__global__ void rem_stream_kernel(const float* __restrict__ ws,
                                  float* __restrict__ out)
{
    __shared__ float tabs[512];
    const int i = blockIdx.x;     // row  0..2047
    const int h = blockIdx.y;     // head 0..7

    // --- CDNA5 async global -> LDS copy of this head's tables (2 KB) -------
    // GVS addressing: saddr (SGPR pair) + per-lane 32-bit VGPR offset.
    // Each of the 256 threads moves 8 bytes -> 2 KB total for the block.
    {
        const unsigned tid  = threadIdx.x;
        const float*   src  = ws + (size_t)h * 512;          // SGPR base
        unsigned       voff = tid * 8u;                      // global byte offset
        unsigned       loff = (unsigned)(uintptr_t)(&tabs[0]) + tid * 8u; // LDS byte addr
        asm volatile("global_load_async_to_lds_b64 %0, %1, %2\n\t"
                     "s_wait_asynccnt 0"
                     :: "v"(loff), "v"(voff), "s"(src)
                     : "memory");
    }
    __syncthreads();

    const float* sTA = tabs;
    const float* sTB = tabs + 256;
    float* outRow = out + (((size_t)h * TRUNC_) + (size_t)i) * TLEN_;

    // dilation factor per head: {1,1,4,1,1,1,3,2}
    const int n = (h == 2) ? 4 : (h == 6) ? 3 : (h == 7) ? 2 : 1;
    switch (n) {
        case 1: emit_row<1>(sTA, sTB, outRow, i); break;
        case 2: emit_row<2>(sTA, sTB, outRow, i); break;
        case 3: emit_row<3>(sTA, sTB, outRow, i); break;
        case 4: emit_row<4>(sTA, sTB, outRow, i); break;
    }
}

extern "C" void kernel_launch(void* const* d_in, const int* in_sizes, int n_in,
                              void* d_out, int out_size, void* d_ws, size_t ws_size,
                              hipStream_t stream)
{
    (void)in_sizes; (void)n_in; (void)out_size; (void)ws_size;
    const float* eta   = (const float*)d_in[0];
    const float* nu    = (const float*)d_in[1];
    const float* theta = (const float*)d_in[2];
    float* ws  = (float*)d_ws;
    float* out = (float*)d_out;

    // 1) one-time table build: 8 heads x 2 tables x 256 entries (16 KB in ws)
    build_tables_kernel<<<dim3(16), dim3(256), 0, stream>>>(eta, nu, theta, ws);

    // 2) bandwidth-bound streaming pass: one block per (row, head), 8 KB/row
    rem_stream_kernel<<<dim3(TRUNC_, NH_), dim3(256), 0, stream>>>(ws, out);
}